// AllRandomBlur_72696616452486
// MI455X (gfx1250) — compile-verified
//
#include <hip/hip_runtime.h>

typedef __attribute__((ext_vector_type(2))) float v2f;
typedef __attribute__((ext_vector_type(8))) float v8f;
typedef __attribute__((ext_vector_type(4))) unsigned int u32x4;
typedef __attribute__((ext_vector_type(8))) int i32x8;
typedef __attribute__((ext_vector_type(4))) int i32x4;

namespace {

constexpr int KS   = 49;   // taps
constexpr int RAD  = 24;   // (KS-1)/2
constexpr int IMH  = 512;
constexpr int IMW  = 512;
constexpr int NC   = 3;
constexpr int WC   = IMW * NC;   // 1536 contiguous floats per (n,h) row
constexpr int NIMG = 64;
constexpr float INV2S2 = 1.0f / 18.0f;   // 1/(2*sigma^2), sigma = 3

// Pass-2 LDS stage: 16 rows x 336 dwords, TDM-padded 1 dword per 16 dwords
// -> logical offset(h, q) = h*357 + q + q/16 ; 357 % 64 = 37 (odd) so the
// 16 lanes reading a column (stride 357) hit 16 distinct banks.
constexpr int STG_Q    = 336;                 // dwords of data per row
constexpr int STG_ROW  = STG_Q + STG_Q / 16;  // 357 dwords incl. padding
constexpr int STG_SIZE = 16 * STG_ROW + 4;    // a little slack for trailing pad

// D = A(16x4) * B(4x16) + C, fp32 matrix pipe
__device__ __forceinline__ v8f wmma4_f32(v2f a, v2f b, v8f c) {
  return __builtin_amdgcn_wmma_f32_16x16x4_f32(
      /*neg_a=*/false, a, /*neg_b=*/false, b,
      /*c_mod=*/(short)0, c, /*reuse_a=*/false, /*reuse_b=*/false);
}

// Normalized 49-tap Gaussian table in LDS (indices 0..48 valid, rest 0).
__device__ __forceinline__ void build_gauss(float* gs, int tid) {
  if (tid < 64) {
    float v = 0.0f;
    if (tid < KS) { float r = (float)tid - (float)RAD; v = __expf(-r * r * INV2S2); }
    gs[tid] = v;
  }
  __syncthreads();
  float s = 0.0f;
  for (int i = 0; i < KS; ++i) s += gs[i];
  __syncthreads();
  if (tid < KS) gs[tid] *= (1.0f / s);
  __syncthreads();
}

// Hoist the 16 distinct banded-A v2f values for this lane into registers.
// A[m,k]: m = ln, K = comp + 2*hx, tap = 4*cr + K - m.
__device__ __forceinline__ void build_areg(const float* gs, int ln, int hx, v2f* Areg) {
  #pragma unroll
  for (int cr = 0; cr < 16; ++cr) {
    const int tx = 4 * cr + 2 * hx - ln;
    const int ty = tx + 1;
    v2f a;
    a.x = ((unsigned)tx <= 48u) ? gs[tx] : 0.0f;
    a.y = ((unsigned)ty <= 48u) ? gs[ty] : 0.0f;
    Areg[cr] = a;
  }
}

// ---------------------------------------------------------------------------
// Pass 1: blur along H. Free axis = contiguous WC axis.
// One wave: 64 output rows (4 stacked 16x16 tiles) x 16 contiguous WC columns.
// B chunks (4 input rows x 16 cols) are loaded once and feed up to 4 tiles.
// ---------------------------------------------------------------------------
__global__ __launch_bounds__(256, 2) void blur_v_wmma(const float* __restrict__ in,
                                                      float* __restrict__ mid) {
  __shared__ float gs[64];
  const int tid = threadIdx.x;
  build_gauss(gs, tid);

  const int lane = tid & 31;
  const int ln   = lane & 15;   // M (A) / N (B,D) index
  const int hx   = lane >> 4;   // half-wave selects K pair / D row group

  const int wave = blockIdx.x * 8 + (tid >> 5);
  const int wcg  = wave % 96;          // 96 column-groups of 16 along WC
  const int hg   = (wave / 96) & 7;    // 8 row-groups of 64 along H
  const int n    = wave / (96 * 8);    // image

  const int wc0   = wcg * 16;
  const int hbase = hg * 64;

  v2f Areg[16];
  build_areg(gs, ln, hx, Areg);

  const float* __restrict__ src = in + (size_t)n * IMH * WC;

  v8f zero = {};
  v8f acc[4] = {zero, zero, zero, zero};

  // 112-row window (hbase-24 .. hbase+87) in 28 chunks of 4 rows
  #pragma unroll
  for (int i = 0; i < 28; ++i) {
    const int jstart = hbase - RAD + 4 * i;
    const int r0 = jstart + 2 * hx;      // K = 0 + 2*hx  (component x)
    const int r1 = r0 + 1;               // K = 1 + 2*hx  (component y)
    v2f b;
    b.x = ((unsigned)r0 < (unsigned)IMH) ? src[(size_t)r0 * WC + wc0 + ln] : 0.0f;
    b.y = ((unsigned)r1 < (unsigned)IMH) ? src[(size_t)r1 * WC + wc0 + ln] : 0.0f;

    #pragma unroll
    for (int t = 0; t < 4; ++t) {
      const int cr = i - 4 * t;                 // chunk index relative to tile t
      if (cr < 0 || cr > 15) continue;          // folds at compile time
      acc[t] = wmma4_f32(Areg[cr], b, acc[t]);
    }
  }

  float* __restrict__ dst = mid + (size_t)n * IMH * WC;
  #pragma unroll
  for (int t = 0; t < 4; ++t) {
    #pragma unroll
    for (int v = 0; v < 8; ++v) {
      const int h = hbase + 16 * t + v + 8 * hx;
      dst[(size_t)h * WC + wc0 + ln] = acc[t][v];
    }
  }
}

// ---------------------------------------------------------------------------
// Pass 2: blur along W. Stage (16h x 112w x 3c) tile into LDS row-major with
// TDM padding (interior tiles: one TENSOR_LOAD_TO_LDS DMA; edge tiles: manual
// staging into the identical padded layout). WMMA per channel reads B columns
// from the padded stage conflict-free; D tiles transpose back through LDS and
// store coalesced. Block = 4 waves, each wave owns 16 w outputs.
// ---------------------------------------------------------------------------
__global__ __launch_bounds__(128, 4) void blur_h_wmma(const float* __restrict__ mid,
                                                      float* __restrict__ out) {
  __shared__ float gs[64];
  __shared__ float st[STG_SIZE];       // padded [hh][q(+pad)] input stage
  __shared__ float ot[16 * 192];       // [hh][w*3+c]  output stage
  const int tid = threadIdx.x;

  const int b  = blockIdx.x;
  const int wb = b & 7;            // 8 w-tiles of 64
  const int hb = (b >> 3) & 31;    // 32 h-tiles of 16
  const int n  = b >> 8;           // image

  const int h0 = hb * 16;
  const int w0 = wb * 64;

  const float* __restrict__ src = mid + ((size_t)n * IMH + h0) * WC;
  const bool interior = (wb >= 1) && (wb <= 6);   // full 112-w window in range

  // Kick off the async TDM tile load first (interior tiles), so the DMA
  // overlaps the Gaussian-table build. One wave issues it (EXEC ignored).
  if (interior && tid < 32) {
    const unsigned lds_off = (unsigned)(size_t)(void*)st;
    const unsigned long long ga =
        (unsigned long long)(size_t)(const void*)(src + (w0 - RAD) * 3);
    u32x4 g0;
    g0[0] = 1u;                                   // count=1, user descriptor
    g0[1] = lds_off;                              // lds_addr
    g0[2] = (unsigned)(ga & 0xFFFFFFFFull);       // global_addr[31:0]
    g0[3] = (unsigned)((ga >> 32) & 0x1FFFFFFull) // global_addr[56:32]
            | (2u << 30);                         // type = 2 ("image")
    i32x8 g1;
    g1[0] = (2 << 16)      // data_size = 4B
          | (1 << 20)      // pad_enable
          | (3 << 22)      // pad_interval: 16 dwords
          | (0 << 25);     // pad_amount: 1 dword
    g1[1] = (STG_Q & 0xFFFF) << 16;               // tensor_dim0[15:0]
    g1[2] = (STG_Q >> 16) | (16 << 16);           // tensor_dim0[31:16], tensor_dim1[15:0]
    g1[3] = (16 >> 16) | (STG_Q << 16);           // tensor_dim1[31:16], tile_dim0
    g1[4] = 16;                                   // tile_dim1 = 16, tile_dim2 = 0
    g1[5] = WC;                                   // tensor_dim0_stride[31:0]
    g1[6] = 0;                                    // stride[47:32], dim1_stride lo
    g1[7] = 0;
    i32x4 gz4 = {0, 0, 0, 0};
    i32x8 gz8 = {0, 0, 0, 0, 0, 0, 0, 0};
    // amdgpu-toolchain (clang-23) 6-arg form; trailing groups disabled/zero.
    __builtin_amdgcn_tensor_load_to_lds(g0, g1, gz4, gz4, gz8, 0);
  }

  build_gauss(gs, tid);

  if (interior) {
    if (tid < 32) __builtin_amdgcn_s_wait_tensorcnt(0);
  } else {
    // Edge tiles: manual zero-padded staging into the same padded layout.
    for (int idx = tid; idx < 16 * STG_Q; idx += 128) {
      const int hh = idx / STG_Q;
      const int q  = idx % STG_Q;                 // q = ww*3 + c
      const int w  = w0 - RAD + q / 3;
      float v = 0.0f;
      if ((unsigned)w < (unsigned)IMW) {
        const int off = (w0 - RAD) * 3 + q;       // == w*3 + c, >= 0 under guard
        v = src[(size_t)hh * WC + off];
      }
      st[hh * STG_ROW + q + (q >> 4)] = v;
    }
  }
  __syncthreads();

  const int lane = tid & 31;
  const int ln   = lane & 15;
  const int hx   = lane >> 4;
  const int wv   = tid >> 5;        // wave id: w outputs [w0+16*wv, +16)

  v2f Areg[16];
  build_areg(gs, ln, hx, Areg);     // A depends only on chunk index i here

  #pragma unroll
  for (int c = 0; c < 3; ++c) {
    v8f acc = {};
    #pragma unroll
    for (int i = 0; i < 16; ++i) {
      const int row = 16 * wv + 4 * i + 2 * hx;   // window-relative w index
      const int qx  = row * 3 + c;
      const int qy  = qx + 3;
      v2f bb;
      bb.x = st[ln * STG_ROW + qx + (qx >> 4)];
      bb.y = st[ln * STG_ROW + qy + (qy >> 4)];
      acc = wmma4_f32(Areg[i], bb, acc);
    }
    #pragma unroll
    for (int v = 0; v < 8; ++v) {
      const int m = v + 8 * hx;                         // w offset in wave tile
      ot[ln * 192 + (16 * wv + m) * 3 + c] = acc[v];    // N lane = h row
    }
  }
  __syncthreads();

  float* __restrict__ dst = out + ((size_t)n * IMH + h0) * WC + (size_t)w0 * 3;
  for (int idx = tid; idx < 16 * 192; idx += 128) {
    dst[(size_t)(idx / 192) * WC + (idx % 192)] = ot[idx];
  }
}

} // namespace

extern "C" void kernel_launch(void* const* d_in, const int* in_sizes, int n_in,
                              void* d_out, int out_size, void* d_ws, size_t ws_size,
                              hipStream_t stream) {
  (void)in_sizes; (void)n_in; (void)out_size; (void)ws_size;
  const float* x  = (const float*)d_in[0];
  float* out      = (float*)d_out;
  float* mid      = (float*)d_ws;   // needs 64*512*512*3*4 B = 192 MiB scratch

  // Pass 1: 64 imgs * 8 h-groups * 96 wc-groups waves, 8 waves/block
  blur_v_wmma<<<(NIMG * 8 * 96) / 8, 256, 0, stream>>>(x, mid);
  // Pass 2: 64 imgs * 32 h-tiles * 8 w-tiles blocks of 128 threads
  blur_h_wmma<<<NIMG * 32 * 8, 128, 0, stream>>>(mid, out);
}